// Attn_58669253263845
// MI455X (gfx1250) — compile-verified
//
#include <hip/hip_runtime.h>
#include <hip/hip_bf16.h>

typedef __attribute__((ext_vector_type(2))) float v2f;
typedef __attribute__((ext_vector_type(8))) float v8f;

#define H    1024
#define SEQ  512
#define BATCH 64

// ---------------------------------------------------------------------------
// Stage 1: u[k] = sum_h v[h] * W[h, 2H + k]   (k in [0,H))
// Implemented with V_WMMA_F32_16X16X4_F32 (f32 WMMA -> exact f32 math).
// One wave per 16-wide column tile of W. v is broadcast into every A row, so
// every row of D equals u's tile; we read row M=0 (VGPR0, lanes 0..15).
// A layout (16x4 f32): lanes 0-15 hold {K0,K1}, lanes 16-31 hold {K2,K3}.
// B layout (4x16 f32): VGPR0 = rows K0(l0-15)/K2(l16-31), VGPR1 = K1/K3.
// ---------------------------------------------------------------------------
__global__ __launch_bounds__(32) void attn_stage1_wmma(
    const float* __restrict__ W,   // [H, 3H] row-major
    const float* __restrict__ v,   // [H]
    float* __restrict__ u)         // [H]
{
    const int lane = threadIdx.x;        // full wave, EXEC all ones
    const int half = lane >> 4;          // 0 or 1
    const int m    = lane & 15;
    const int col  = 2 * H + blockIdx.x * 16 + m;   // W column for this lane

    v8f c = {};
    for (int h = 0; h < H; h += 4) {
        const int k0 = h + 2 * half;     // K index pair handled by this half
        v2f a, bm;
        a.x  = v[k0];
        a.y  = v[k0 + 1];
        bm.x = W[(size_t)k0 * (3 * H) + col];
        bm.y = W[(size_t)(k0 + 1) * (3 * H) + col];
        // 8 args: (neg_a, A, neg_b, B, c_mod, C, reuse_a, reuse_b)
        c = __builtin_amdgcn_wmma_f32_16x16x4_f32(
                false, a, false, bm, (short)0, c, false, false);
    }
    if (lane < 16) u[blockIdx.x * 16 + lane] = c[0];   // row M=0 of D
}

// ---------------------------------------------------------------------------
// Stage 2: scores[row] = dot(enc[row, :], u)   row = b*SEQ + s  (32768 rows)
// Bandwidth-bound streaming pass: one wave per row, float4 (b128) loads,
// wave32 shuffle tree reduction. 8 waves per 256-thread block.
// ---------------------------------------------------------------------------
__global__ __launch_bounds__(256) void attn_scores(
    const float* __restrict__ enc,     // [B*S, H]
    const float* __restrict__ u,       // [H] (stays hot in cache)
    float* __restrict__ scores)        // [B*S]
{
    const int wave = threadIdx.x >> 5;
    const int lane = threadIdx.x & 31;
    const size_t row = (size_t)blockIdx.x * 8 + wave;

    const float4* e  = (const float4*)(enc + row * H);
    const float4* uv = (const float4*)u;

    float acc = 0.0f;
#pragma unroll
    for (int i = 0; i < H / (32 * 4); ++i) {   // 8 iterations
        float4 ev = e[i * 32 + lane];
        float4 us = uv[i * 32 + lane];
        acc = fmaf(ev.x, us.x, acc);
        acc = fmaf(ev.y, us.y, acc);
        acc = fmaf(ev.z, us.z, acc);
        acc = fmaf(ev.w, us.w, acc);
    }
#pragma unroll
    for (int off = 16; off > 0; off >>= 1)
        acc += __shfl_xor(acc, off, 32);

    if (lane == 0) scores[row] = acc;
}

// ---------------------------------------------------------------------------
// Stage 3: out[b, :] = softmax(scores[b, :]) over SEQ. One block per b,
// 512 threads (16 wave32s), shuffle + LDS two-level reductions.
// ---------------------------------------------------------------------------
__global__ __launch_bounds__(512) void attn_softmax(
    const float* __restrict__ scores, float* __restrict__ out)
{
    __shared__ float red[16];
    const int b    = blockIdx.x;
    const int t    = threadIdx.x;
    const int lane = t & 31;
    const int wave = t >> 5;

    const float sc = scores[b * SEQ + t];

    // --- max ---
    float m = sc;
#pragma unroll
    for (int off = 16; off > 0; off >>= 1)
        m = fmaxf(m, __shfl_xor(m, off, 32));
    if (lane == 0) red[wave] = m;
    __syncthreads();
    if (wave == 0) {
        float mm = red[lane & 15];
#pragma unroll
        for (int off = 8; off > 0; off >>= 1)
            mm = fmaxf(mm, __shfl_xor(mm, off, 32));
        if (lane == 0) red[0] = mm;
    }
    __syncthreads();
    m = red[0];
    __syncthreads();

    // --- sum of exp ---
    const float e = expf(sc - m);
    float s = e;
#pragma unroll
    for (int off = 16; off > 0; off >>= 1)
        s += __shfl_xor(s, off, 32);
    if (lane == 0) red[wave] = s;
    __syncthreads();
    if (wave == 0) {
        float ss = red[lane & 15];
#pragma unroll
        for (int off = 8; off > 0; off >>= 1)
            ss += __shfl_xor(ss, off, 32);
        if (lane == 0) red[0] = ss;
    }
    __syncthreads();

    out[b * SEQ + t] = e / red[0];
}

// ---------------------------------------------------------------------------
// Inputs (setup_inputs order): 0=hidden [2,B,H], 1=encoder_outputs [B,S,H],
// 2=W [H,3H], 3=b [H], 4=v [H].  hidden, b, W[:,0:2H] drop out of the softmax
// (additive constants along the softmax axis).
// ---------------------------------------------------------------------------
extern "C" void kernel_launch(void* const* d_in, const int* in_sizes, int n_in,
                              void* d_out, int out_size, void* d_ws, size_t ws_size,
                              hipStream_t stream) {
    const float* enc = (const float*)d_in[1];
    const float* W   = (const float*)d_in[2];
    const float* v   = (const float*)d_in[4];
    float* out = (float*)d_out;

    float* u      = (float*)d_ws;   // H floats
    float* scores = u + H;          // BATCH*SEQ floats

    attn_stage1_wmma<<<H / 16, 32, 0, stream>>>(W, v, u);
    attn_scores<<<(BATCH * SEQ) / 8, 256, 0, stream>>>(enc, u, scores);
    attn_softmax<<<BATCH, SEQ, 0, stream>>>(scores, out);
}